// augGNN_23991687315868
// MI455X (gfx1250) — compile-verified
//
#include <hip/hip_runtime.h>
#include <math.h>

typedef __attribute__((ext_vector_type(2))) float v2f;
typedef __attribute__((ext_vector_type(8))) float v8f;

#define BN_INV 0.9999950000374997f  /* 1/sqrt(1+1e-5) */

// ---------------------------------------------------------------------------
// K1a: s[i] = x[i, col]
__global__ void k_init_s(const float* __restrict__ x, float* __restrict__ s,
                         int N, int col) {
  int i = blockIdx.x * blockDim.x + threadIdx.x;
  if (i < N) s[i] = x[2 * i + col];
}

// K1b: s[dst] += x[src, col] over all edges (scalar segment-sum)
__global__ void k_agg_s(const float* __restrict__ x, const int* __restrict__ ei,
                        float* __restrict__ s, int E, int col) {
  int e = blockIdx.x * blockDim.x + threadIdx.x;
  if (e < E) {
    int src = ei[e];
    int dst = ei[E + e];
    atomicAdd(&s[dst], x[2 * src + col]);
  }
}

// ---------------------------------------------------------------------------
// K2: H1 = relu(bn1(s ⊗ W1 + b1)) @ W2 + b2    [N,128], K=256, fp32 WMMA.
// A is generated on the fly: A[i,k] = relu(s_i*gw[k] + gb[k]) where
// gw[k] = g1[k]*inv*W1[k], gb[k] = g1[k]*inv*b1[k] + be1[k].
// Workgroup: 16 rows x 128 cols; 8 waves, each one 16x16 tile.
__global__ void __launch_bounds__(256) k_gin1_gemm(
    const float* __restrict__ s, const float* __restrict__ W1,
    const float* __restrict__ b1, const float* __restrict__ g1,
    const float* __restrict__ be1, const float* __restrict__ W2,
    const float* __restrict__ b2, float* __restrict__ H1, int N) {
  __shared__ float gw[256];
  __shared__ float gb[256];
  int tid = threadIdx.x;
  {
    float g = g1[tid] * BN_INV;
    gw[tid] = g * W1[tid];
    gb[tid] = fmaf(g, b1[tid], be1[tid]);
  }
  __syncthreads();

  int wave = tid >> 5;
  int lane = tid & 31;
  int half = lane >> 4;   // 0: K pair {k,k+1}, 1: {k+2,k+3}
  int l16  = lane & 15;
  int rowBase = blockIdx.x * 16;
  int colBase = wave * 16;
  int row = rowBase + l16;
  if (row >= N) row = N - 1;            // clamp: keep EXEC all-1s for WMMA
  float sv = s[row];
  int col = colBase + l16;

  v8f c = {};
  #pragma unroll 4
  for (int k0 = 0; k0 < 256; k0 += 4) {
    int kb = k0 + (half << 1);
    v2f a, b;
    a.x = fmaxf(fmaf(sv, gw[kb],     gb[kb]),     0.f);
    a.y = fmaxf(fmaf(sv, gw[kb + 1], gb[kb + 1]), 0.f);
    b.x = W2[kb * 128 + col];
    b.y = W2[(kb + 1) * 128 + col];
    c = __builtin_amdgcn_wmma_f32_16x16x4_f32(false, a, false, b,
                                              (short)0, c, false, false);
  }
  float bb = b2[col];
  #pragma unroll
  for (int j = 0; j < 8; ++j) {
    int m = j + (half << 3);            // C/D: VGPR j -> M=j (+8 for hi half)
    int r = rowBase + m;
    if (r < N) H1[(long)r * 128 + col] = c[j] + bb;
  }
}

// ---------------------------------------------------------------------------
// K3a: Z2 = H1 (float4 copy)
__global__ void k_copy4(const float4* __restrict__ src, float4* __restrict__ dst,
                        long n4) {
  long i = (long)blockIdx.x * blockDim.x + threadIdx.x;
  if (i < n4) dst[i] = src[i];
}

// K3b: Z2[dst,:] += H1[src,:]  (one wave of 32 lanes per edge, 4 rounds of 32)
__global__ void __launch_bounds__(256) k_agg_feat(
    const float* __restrict__ H1, const int* __restrict__ ei,
    float* __restrict__ Z2, int E) {
  int g = blockIdx.x * 8 + (threadIdx.x >> 5);
  if (g >= E) return;
  int lane = threadIdx.x & 31;
  int src = ei[g];
  int dst = ei[E + g];
  const float* hs = H1 + (long)src * 128;
  float* zd = Z2 + (long)dst * 128;
  #pragma unroll
  for (int r = 0; r < 4; ++r) {
    int f = lane + 32 * r;
    atomicAdd(&zd[f], hs[f]);
  }
}

// ---------------------------------------------------------------------------
// K4: H2 = relu(bn2(Z2 @ W3 + b3)) @ W4 + b4   [N,64], both GEMMs fused.
// Workgroup covers 32 rows; intermediate 32x64 tile staged in padded LDS.
__global__ void __launch_bounds__(256) k_gin2_fused(
    const float* __restrict__ Z2, const float* __restrict__ W3,
    const float* __restrict__ b3, const float* __restrict__ g2,
    const float* __restrict__ be2, const float* __restrict__ W4,
    const float* __restrict__ b4, float* __restrict__ H2, int N) {
  __shared__ float T[32 * 68];          // stride 68 floats: avoid bank conflicts
  int tid  = threadIdx.x;
  int wave = tid >> 5;
  int lane = tid & 31;
  int half = lane >> 4;
  int l16  = lane & 15;
  int rtl     = wave >> 2;              // row tile 0..1
  int colBase = (wave & 3) * 16;        // col tile 0..3
  int rowBase = blockIdx.x * 32;
  int row  = rowBase + rtl * 16 + l16;
  int rowC = row < N ? row : N - 1;
  int col  = colBase + l16;
  const float* zrow = Z2 + (long)rowC * 128;

  // GEMM 1: Z2(16x128) @ W3(128x64)
  v8f c = {};
  #pragma unroll 4
  for (int k0 = 0; k0 < 128; k0 += 4) {
    int kb = k0 + (half << 1);
    v2f a, b;
    a.x = zrow[kb];
    a.y = zrow[kb + 1];
    b.x = W3[kb * 64 + col];
    b.y = W3[(kb + 1) * 64 + col];
    c = __builtin_amdgcn_wmma_f32_16x16x4_f32(false, a, false, b,
                                              (short)0, c, false, false);
  }
  // epilogue 1: +b3, bn2, relu -> LDS tile
  float gc = g2[col] * BN_INV;
  float bc = b3[col];
  float ec = be2[col];
  #pragma unroll
  for (int j = 0; j < 8; ++j) {
    int m = j + (half << 3);
    float v = c[j] + bc;
    v = fmaxf(fmaf(gc, v, ec), 0.f);
    T[(rtl * 16 + m) * 68 + col] = v;
  }
  __syncthreads();

  // GEMM 2: T(16x64) @ W4(64x64)
  v8f d = {};
  const float* trow = &T[(rtl * 16 + l16) * 68];
  #pragma unroll 4
  for (int k0 = 0; k0 < 64; k0 += 4) {
    int kb = k0 + (half << 1);
    v2f a, b;
    a.x = trow[kb];
    a.y = trow[kb + 1];
    b.x = W4[kb * 64 + col];
    b.y = W4[(kb + 1) * 64 + col];
    d = __builtin_amdgcn_wmma_f32_16x16x4_f32(false, a, false, b,
                                              (short)0, d, false, false);
  }
  float b4c = b4[col];
  #pragma unroll
  for (int j = 0; j < 8; ++j) {
    int m = j + (half << 3);
    int r = rowBase + rtl * 16 + m;
    if (r < N) H2[(long)r * 64 + col] = d[j] + b4c;
  }
}

// ---------------------------------------------------------------------------
// K5: head: relu(concat(e1,e2) @ Wl1 + bl1) @ Wl2 + bl2, then log_softmax.
__global__ void __launch_bounds__(256) k_head(
    const float* __restrict__ H2, const float* __restrict__ Wl1,
    const float* __restrict__ bl1, const float* __restrict__ Wl2,
    const float* __restrict__ bl2, float* __restrict__ out, int N) {
  __shared__ float w1s[128 * 6];
  __shared__ float w2s[36];
  __shared__ float b1s[6];
  __shared__ float b2s[6];
  for (int i = threadIdx.x; i < 768; i += 256) w1s[i] = Wl1[i];
  if (threadIdx.x < 36) w2s[threadIdx.x] = Wl2[threadIdx.x];
  if (threadIdx.x < 6) {
    b1s[threadIdx.x] = bl1[threadIdx.x];
    b2s[threadIdx.x] = bl2[threadIdx.x];
  }
  __syncthreads();
  int i = blockIdx.x * blockDim.x + threadIdx.x;
  if (i >= N) return;

  float h[6];
  #pragma unroll
  for (int j = 0; j < 6; ++j) h[j] = b1s[j];
  const float* e1 = H2 + (long)i * 64;
  const float* e2 = H2 + (long)(N + i) * 64;
  for (int k = 0; k < 64; ++k) {
    float v = e1[k];
    #pragma unroll
    for (int j = 0; j < 6; ++j) h[j] = fmaf(v, w1s[k * 6 + j], h[j]);
  }
  for (int k = 0; k < 64; ++k) {
    float v = e2[k];
    #pragma unroll
    for (int j = 0; j < 6; ++j) h[j] = fmaf(v, w1s[(64 + k) * 6 + j], h[j]);
  }
  #pragma unroll
  for (int j = 0; j < 6; ++j) h[j] = fmaxf(h[j], 0.f);

  float g[6];
  #pragma unroll
  for (int j = 0; j < 6; ++j) {
    float acc = b2s[j];
    #pragma unroll
    for (int k = 0; k < 6; ++k) acc = fmaf(h[k], w2s[k * 6 + j], acc);
    g[j] = acc;
  }
  float m = g[0];
  #pragma unroll
  for (int j = 1; j < 6; ++j) m = fmaxf(m, g[j]);
  float sum = 0.f;
  #pragma unroll
  for (int j = 0; j < 6; ++j) sum += __expf(g[j] - m);
  float lse = m + __logf(sum);
  #pragma unroll
  for (int j = 0; j < 6; ++j) out[(long)i * 6 + j] = g[j] - lse;
}

// ---------------------------------------------------------------------------
extern "C" void kernel_launch(void* const* d_in, const int* in_sizes, int n_in,
                              void* d_out, int out_size, void* d_ws,
                              size_t ws_size, hipStream_t stream) {
  const float* x   = (const float*)d_in[0];
  const int*   ei  = (const int*)d_in[1];
  const float* W1  = (const float*)d_in[2];
  const float* b1  = (const float*)d_in[3];
  const float* g1  = (const float*)d_in[4];
  const float* be1 = (const float*)d_in[5];
  const float* W2  = (const float*)d_in[6];
  const float* b2  = (const float*)d_in[7];
  const float* W3  = (const float*)d_in[8];
  const float* b3  = (const float*)d_in[9];
  const float* g2  = (const float*)d_in[10];
  const float* be2 = (const float*)d_in[11];
  const float* W4  = (const float*)d_in[12];
  const float* b4  = (const float*)d_in[13];
  const float* Wl1 = (const float*)d_in[14];
  const float* bl1 = (const float*)d_in[15];
  const float* Wl2 = (const float*)d_in[16];
  const float* bl2 = (const float*)d_in[17];
  float* out = (float*)d_out;

  int N = in_sizes[0] / 2;
  int E = in_sizes[1] / 2;
  (void)n_in; (void)out_size; (void)ws_size;

  float* ws = (float*)d_ws;
  size_t off = 0;
  float* s  = ws + off; off += (size_t)N;        off = (off + 3) & ~(size_t)3;
  float* H1 = ws + off; off += (size_t)N * 128;
  float* Z2 = ws + off; off += (size_t)N * 128;
  float* H2 = ws + off; off += (size_t)N * 128;  // both passes: 2 * N*64

  for (int pass = 0; pass < 2; ++pass) {
    int colsel = (pass == 0) ? 1 : 0;            // column 1 first, then column 0
    float* h2out = H2 + (size_t)pass * (size_t)N * 64;

    k_init_s<<<(N + 255) / 256, 256, 0, stream>>>(x, s, N, colsel);
    k_agg_s<<<(E + 255) / 256, 256, 0, stream>>>(x, ei, s, E, colsel);
    k_gin1_gemm<<<(N + 15) / 16, 256, 0, stream>>>(s, W1, b1, g1, be1, W2, b2,
                                                   H1, N);
    long n4 = (long)N * 32;
    k_copy4<<<(int)((n4 + 255) / 256), 256, 0, stream>>>((const float4*)H1,
                                                         (float4*)Z2, n4);
    k_agg_feat<<<(E + 7) / 8, 256, 0, stream>>>(H1, ei, Z2, E);
    k_gin2_fused<<<(N + 31) / 32, 256, 0, stream>>>(Z2, W3, b3, g2, be2, W4,
                                                    b4, h2out, N);
  }
  k_head<<<(N + 255) / 256, 256, 0, stream>>>(H2, Wl1, bl1, Wl2, bl2, out, N);
}